// LSH_26328149524717
// MI455X (gfx1250) — compile-verified
//
#include <hip/hip_runtime.h>
#include <hip/hip_bf16.h>

typedef float v2f __attribute__((ext_vector_type(2)));
typedef float v8f __attribute__((ext_vector_type(8)));

#define TBL_BITS 19
#define TBL_SIZE (1u << TBL_BITS)
#define EMPTY_KEY ((int)0x80000000)   // INT_MIN sentinel; |bucket| <= ~5e4, never collides
#define VAL_INIT  0x7FFFFFFF

__device__ __forceinline__ unsigned hash32(int k) {
    return (unsigned)k * 2654435761u;
}

__device__ __forceinline__ void table_insert(int* __restrict__ keys,
                                             int* __restrict__ vals,
                                             int key, int idx) {
    unsigned slot = hash32(key) & (TBL_SIZE - 1);
    for (;;) {
        int prev = atomicCAS(&keys[slot], EMPTY_KEY, key);
        if (prev == EMPTY_KEY || prev == key) {
            atomicMin(&vals[slot], idx);
            return;
        }
        slot = (slot + 1) & (TBL_SIZE - 1);
    }
}

__device__ __forceinline__ int table_lookup(const int* __restrict__ keys,
                                            const int* __restrict__ vals,
                                            int key) {
    unsigned slot = hash32(key) & (TBL_SIZE - 1);
    for (;;) {
        int k = keys[slot];
        if (k == key) return vals[slot];
        if (k == EMPTY_KEY) return 0;   // unreachable: every key was inserted
        slot = (slot + 1) & (TBL_SIZE - 1);
    }
}

__global__ void lsh_table_init(int* __restrict__ keys, int* __restrict__ vals) {
    unsigned i = blockIdx.x * blockDim.x + threadIdx.x;
    if (i < TBL_SIZE) {
        keys[i] = EMPTY_KEY;
        vals[i] = VAL_INIT;
    }
}

// One wave computes 16 f32 dot products dot(x_row, w) via chained
// V_WMMA_F32_16X16X4_F32 (A = 16x4 tile of x, B = w replicated across N),
// then lanes 0 / 16 truncate-to-int and insert (bucket, row) into the table.
__global__ void lsh_dot_insert(const float* __restrict__ x,
                               const float* __restrict__ w,
                               int* __restrict__ v,
                               int* __restrict__ keys,
                               int* __restrict__ vals,
                               int B, int D) {
    const int lane = threadIdx.x & 31;
    const int wave = (blockIdx.x * blockDim.x + threadIdx.x) >> 5;
    const long base = (long)wave * 16;
    if (base >= B) return;                 // uniform per wave: EXEC stays all-1s

    const int m    = lane & 15;            // A-matrix row this lane carries
    const int koff = (lane >> 4) * 2;      // K group: lanes 0-15 -> K=0,1 ; 16-31 -> K=2,3
    long row = base + m;
    if (row >= B) row = B - 1;             // clamp partial tile (keeps wave full)

    const float* xr = x + row * (long)D + koff;
    const float* wr = w + koff;

    v8f c = {};
    for (int k = 0; k < D; k += 4) {
        v2f a = *(const v2f*)(xr + k);     // A[m][k+koff .. k+koff+1]
        v2f b = *(const v2f*)(wr + k);     // B[k][*] = w[k] replicated across N
        c = __builtin_amdgcn_wmma_f32_16x16x4_f32(
                /*neg_a=*/false, a, /*neg_b=*/false, b,
                /*c_mod=*/(short)0, c, /*reuse_a=*/false, /*reuse_b=*/false);
    }

    // C/D layout: lane 0 holds rows base+0..7 in c[0..7]; lane 16 holds base+8..15.
    if ((lane & 15) == 0) {
        const int rbase = (lane >> 4) * 8;
#pragma unroll
        for (int r = 0; r < 8; ++r) {
            long rr = base + rbase + r;
            if (rr < B) {
                int key = (int)truncf(c[r]);
                v[rr] = key;
                table_insert(keys, vals, key, (int)rr);
            }
        }
    }
}

__global__ void lsh_resolve(const int* __restrict__ v,
                            int* __restrict__ first,
                            const int* __restrict__ keys,
                            const int* __restrict__ vals,
                            int B) {
    int i = blockIdx.x * blockDim.x + threadIdx.x;
    if (i < B) first[i] = table_lookup(keys, vals, v[i]);
}

// One wave per output row: 32 lanes x float4 = 512B coalesced copy.
__global__ void lsh_gather(const float* __restrict__ x,
                           const int* __restrict__ first,
                           float* __restrict__ out,
                           int B, int D) {
    const int lane = threadIdx.x & 31;
    const long row = (long)blockIdx.x * (blockDim.x >> 5) + (threadIdx.x >> 5);
    if (row >= B) return;
    const int f = first[row];
    const float4* __restrict__ src = (const float4*)(x + (long)f * D);
    float4* __restrict__ dst = (float4*)(out + row * (long)D);
    for (int t = lane; t < (D >> 2); t += 32) dst[t] = src[t];
}

extern "C" void kernel_launch(void* const* d_in, const int* in_sizes, int n_in,
                              void* d_out, int out_size, void* d_ws, size_t ws_size,
                              hipStream_t stream) {
    const float* x = (const float*)d_in[0];
    const float* w = (const float*)d_in[1];
    float* out = (float*)d_out;

    const int D = in_sizes[1];             // 128
    const int B = in_sizes[0] / D;         // 262144

    // Workspace layout: v[B] | first[B] | keys[TBL] | vals[TBL]  (~6.3 MB)
    int* v     = (int*)d_ws;
    int* first = v + B;
    int* keys  = first + B;
    int* vals  = keys + TBL_SIZE;

    lsh_table_init<<<(TBL_SIZE + 255) / 256, 256, 0, stream>>>(keys, vals);

    const int waves  = (B + 15) / 16;
    const int thr    = waves * 32;
    lsh_dot_insert<<<(thr + 255) / 256, 256, 0, stream>>>(x, w, v, keys, vals, B, D);

    lsh_resolve<<<(B + 255) / 256, 256, 0, stream>>>(v, first, keys, vals, B);

    const long gthr = (long)B * 32;
    lsh_gather<<<(int)((gthr + 255) / 256), 256, 0, stream>>>(x, first, out, B, D);
}